// dual_softmax_loss_90194313216570
// MI455X (gfx1250) — compile-verified
//
#include <hip/hip_runtime.h>
#include <math.h>

// Problem constants
constexpr int N   = 8192;
constexpr int TPB = 256;   // 8 wave32 waves per block

// 16-byte vector type matching the async-to-LDS builtin's parameter type
typedef int v4i __attribute__((vector_size(16)));

// -------------------------------------------------------------------------
// Online (max, sum-exp) update for one value.
__device__ __forceinline__ void online_update(float& M, float& L, float v) {
    float nm = fmaxf(M, v);
    L = L * __expf(M - nm) + __expf(v - nm);
    M = nm;
}

// -------------------------------------------------------------------------
// Kernel 1: partial column-wise softmax stats.
// grid = (N/(4*TPB), ch); each thread owns 4 consecutive columns (float4),
// scans `rpc` rows of its chunk. Loads are 16B/lane, fully coalesced.
__global__ __launch_bounds__(TPB) void dsl_colstats(
    const float* __restrict__ S, const int* __restrict__ tempp,
    float* __restrict__ pmax, float* __restrict__ psum, int rpc) {
    const float invT = 1.0f / (float)(*tempp);
    const int c4   = blockIdx.x * TPB + threadIdx.x;     // float4 column index
    const int row0 = blockIdx.y * rpc;
    const float4* S4 = (const float4*)S;

    float4 M = make_float4(-INFINITY, -INFINITY, -INFINITY, -INFINITY);
    float4 L = make_float4(0.f, 0.f, 0.f, 0.f);

    size_t idx = (size_t)row0 * (N / 4) + c4;
    for (int r = 0; r < rpc; ++r, idx += N / 4) {
        float4 x = S4[idx];
        online_update(M.x, L.x, x.x * invT);
        online_update(M.y, L.y, x.y * invT);
        online_update(M.z, L.z, x.z * invT);
        online_update(M.w, L.w, x.w * invT);
    }
    ((float4*)pmax)[(size_t)blockIdx.y * (N / 4) + c4] = M;
    ((float4*)psum)[(size_t)blockIdx.y * (N / 4) + c4] = L;
}

// -------------------------------------------------------------------------
// Kernel 2: fold ch partials per column into c_j = m_j + log Z_j - log N.
__global__ __launch_bounds__(TPB) void dsl_combine(
    const float* __restrict__ pmax, const float* __restrict__ psum,
    float* __restrict__ c, int ch) {
    const int j = blockIdx.x * TPB + threadIdx.x;
    float m = -INFINITY;
    for (int k = 0; k < ch; ++k) m = fmaxf(m, pmax[(size_t)k * N + j]);
    float Z = 0.f;
    for (int k = 0; k < ch; ++k)
        Z += psum[(size_t)k * N + j] * __expf(pmax[(size_t)k * N + j] - m);
    c[j] = m + logf(Z) - logf((float)N);
}

// -------------------------------------------------------------------------
// Kernel 3: one block per row. Stage the 32KB row into LDS with the CDNA5
// async global->LDS data mover, compute r_ij = x*exp(x/T - c_j) in place,
// then row logsumexp via wave32 shuffles + cross-wave LDS reduce.
__global__ __launch_bounds__(TPB) void dsl_rowloss(
    const float* __restrict__ S, const float* __restrict__ c,
    const int* __restrict__ tempp, float* __restrict__ out) {
    __shared__ __align__(16) float tile[N];   // 32 KB row tile (reused for r)
    __shared__ float red[TPB / 32];
    __shared__ float diag;

    const int row = blockIdx.x;
    const int tid = threadIdx.x;
    const float invT = 1.0f / (float)(*tempp);
    const float* rowp = S + (size_t)row * N;

    // ---- stage row -> LDS via async tensor path (ASYNCcnt-tracked) --------
    for (int k = 0; k < N / 4 / TPB; ++k) {            // 8 x b128 per lane
        int q = k * TPB + tid;                          // float4 index
        __builtin_amdgcn_global_load_async_to_lds_b128(
            (v4i*)(rowp + 4 * q),
            (__attribute__((address_space(3))) v4i*)(&tile[4 * q]),
            /*offset=*/0, /*cpol=*/0);
    }
#if __has_builtin(__builtin_amdgcn_s_wait_asynccnt)
    __builtin_amdgcn_s_wait_asynccnt(0);
#else
    asm volatile("s_wait_asynccnt 0" ::: "memory");
#endif
    __syncthreads();

    // ---- phase A: r = x * exp(x/T - c_j), in place; track local max -------
    float lmax = -INFINITY;
    const float4* c4 = (const float4*)c;
    for (int k = 0; k < N / 4 / TPB; ++k) {
        int q = k * TPB + tid;
        float4 x  = ((float4*)tile)[q];
        float4 cc = c4[q];                 // 32KB vector, WGP$/L2 resident
        float4 r;
        r.x = x.x * __expf(x.x * invT - cc.x);
        r.y = x.y * __expf(x.y * invT - cc.y);
        r.z = x.z * __expf(x.z * invT - cc.z);
        r.w = x.w * __expf(x.w * invT - cc.w);
        lmax = fmaxf(lmax, fmaxf(fmaxf(r.x, r.y), fmaxf(r.z, r.w)));
        int j0 = 4 * q;
        int d = row - j0;
        if (d >= 0 && d < 4)
            diag = (d == 0) ? r.x : (d == 1) ? r.y : (d == 2) ? r.z : r.w;
        ((float4*)tile)[q] = r;            // own slots only: race-free
    }

    // ---- reduce max: wave32 shuffle then cross-wave via LDS ---------------
    float m = lmax;
    for (int off = 16; off > 0; off >>= 1) m = fmaxf(m, __shfl_xor(m, off));
    if ((tid & 31) == 0) red[tid >> 5] = m;
    __syncthreads();
    float M = red[0];
    for (int w = 1; w < TPB / 32; ++w) M = fmaxf(M, red[w]);
    __syncthreads();                       // red reused below

    // ---- phase B: sum exp(r - M) ------------------------------------------
    float ls = 0.f;
    for (int k = 0; k < N / 4 / TPB; ++k) {
        int q = k * TPB + tid;
        float4 r = ((float4*)tile)[q];
        ls += __expf(r.x - M) + __expf(r.y - M) +
              __expf(r.z - M) + __expf(r.w - M);
    }
    for (int off = 16; off > 0; off >>= 1) ls += __shfl_xor(ls, off);
    if ((tid & 31) == 0) red[tid >> 5] = ls;
    __syncthreads();

    if (tid == 0) {
        float s = 0.f;
        for (int w = 0; w < TPB / 32; ++w) s += red[w];
        out[row] = M + logf(s) - diag;     // loss_i = lse_j(r) - r_ii
    }
}

// -------------------------------------------------------------------------
extern "C" void kernel_launch(void* const* d_in, const int* in_sizes, int n_in,
                              void* d_out, int out_size, void* d_ws, size_t ws_size,
                              hipStream_t stream) {
    const float* S    = (const float*)d_in[0];
    const int*   temp = (const int*)d_in[1];
    float*       out  = (float*)d_out;

    // Pick chunk count (pass-1 row split) that fits the workspace:
    // ws layout: pmax[ch*N] | psum[ch*N] | c[N]
    int ch = 128;                                      // 1024 pass-1 blocks
    while (ch > 1 && (size_t)(2 * ch + 1) * N * sizeof(float) > ws_size) ch >>= 1;
    const int rpc = N / ch;

    float* pmax = (float*)d_ws;
    float* psum = pmax + (size_t)ch * N;
    float* cvec = psum + (size_t)ch * N;

    dim3 g1(N / (4 * TPB), ch);
    dsl_colstats<<<g1, TPB, 0, stream>>>(S, temp, pmax, psum, rpc);
    dsl_combine<<<N / TPB, TPB, 0, stream>>>(pmax, psum, cvec, ch);
    dsl_rowloss<<<N, TPB, 0, stream>>>(S, cvec, temp, out);
}